// DTWDistance_19834158973693
// MI455X (gfx1250) — compile-verified
//
#include <hip/hip_runtime.h>
#include <hip/hip_bf16.h>
#include <stdint.h>

// DTW (symmetric2, L1 cost) over 64 independent 1024x1024 DPs.
// One workgroup per batch element; anti-diagonal wavefront in LDS.
// WMMA is inapplicable (min-plus semiring); CDNA5 paths used instead:
//   - global_load_async_to_lds_b128 + s_wait_asynccnt (async staging of x,y)
//   - split workgroup barriers (s_barrier_signal/-wait via __syncthreads)
//
// Fully branchless DP: shifted diagonal buffers pre-filled with HUGE make the
// uniform recurrence min(diag+2d, min(up,left)+d) absorb the i==0/j==0
// borders; off-diagonal lanes store harmless quasi-HUGE values (proof in
// comments), so the inner loop has no compares, selects, or exec-mask ops.
// The k-loop is unrolled by 3 so buffer bases become immediate DS offsets and
// the per-column address VGPRs are loop-invariant.

#define DTW_N 1024
#define DTW_T 256            // threads per block (8 wave32)
#define DTW_Q (DTW_N / DTW_T)
#define DTW_W (DTW_N + 1)    // shifted diagonal buffer width
#define DTW_HUGE 1.0e30f

__global__ __launch_bounds__(DTW_T) void dtw_wavefront_kernel(
    const float* __restrict__ X,       // [B, N]
    const float* __restrict__ Y,       // [B, N]
    float* __restrict__ per_batch)     // [B] normalized distances
{
    __shared__ __align__(16) float sx[DTW_N];
    __shared__ __align__(16) float sy[DTW_N];
    __shared__ __align__(16) float buf0[DTW_W];
    __shared__ __align__(16) float buf1[DTW_W];
    __shared__ __align__(16) float buf2[DTW_W];

    const int b = blockIdx.x;
    const int t = threadIdx.x;

    const float* xg = X + (size_t)b * DTW_N;
    const float* yg = Y + (size_t)b * DTW_N;

    // ---- Stage x,y into LDS via CDNA5 async-to-LDS DMA path ----
    // 256 threads x 16B per array (256*16 = 4096 B = N floats each).
    {
        // Low 32 bits of the generic pointer to a __shared__ object are the
        // LDS byte offset (LDS aperture truncates to addr[31:0]).
        unsigned lds_x = (unsigned)(uintptr_t)(&sx[0]) + (unsigned)(t * 16);
        unsigned lds_y = (unsigned)(uintptr_t)(&sy[0]) + (unsigned)(t * 16);
        unsigned goff  = (unsigned)(t * 16);
        // GVS mode: 64-bit SGPR base + 32-bit per-lane VGPR offset.
        asm volatile("global_load_async_to_lds_b128 %0, %1, %2"
                     :
                     : "v"(lds_x), "v"(goff), "s"(xg)
                     : "memory");
        asm volatile("global_load_async_to_lds_b128 %0, %1, %2"
                     :
                     : "v"(lds_y), "v"(goff), "s"(yg)
                     : "memory");
#if defined(__has_builtin) && __has_builtin(__builtin_amdgcn_s_wait_asynccnt)
        __builtin_amdgcn_s_wait_asynccnt(0);
#else
        asm volatile("s_wait_asynccnt 0" ::: "memory");
#endif
    }

    // ---- Initialize diagonal buffers to HUGE; fold in the (0,0) seed ----
    // (async data already complete for this wave, so reading sx[0]/sy[0] by a
    // lane of wave 0 is safe without a workgroup barrier)
#pragma unroll
    for (int q = 0; q < DTW_Q; ++q) {
        const int p = t + q * DTW_T;
        const float seed = fabsf(sx[0] - sy[0]);     // D[0,0] = d(0,0)
        const float v0 = (p == 1) ? seed : DTW_HUGE; // seed lives at buf0[1]
        buf0[p] = v0;
        buf1[p] = DTW_HUGE;
        buf2[p] = DTW_HUGE;
    }
    if (t == 0) {
        buf0[DTW_N] = DTW_HUGE;
        buf1[DTW_N] = DTW_HUGE;
        buf2[DTW_N] = DTW_HUGE;
    }
    __syncthreads();

    // sy[j] is k-invariant for this thread's columns: keep it in registers.
    float syr[DTW_Q];
#pragma unroll
    for (int q = 0; q < DTW_Q; ++q) syr[q] = sy[t + q * DTW_T];

    // ---- One anti-diagonal step (uniform, branchless, unconditional) ----
    // Cell (i,j) of diagonal k stored at position j+1:
    //   left = D[i,j-1] -> p1[j]
    //   up   = D[i-1,j] -> p1[j+1]
    //   diag = D[i-1,j-1] -> p2[j]
    // Off-diagonal lanes (i<0 or i>N-1) store quasi-HUGE / dead values:
    //   i<0   : inputs are all >=~1e30, store ~1e30 (acts as +inf later)
    //   i>N-1 : positions <= jlo are never read again before overwrite
    auto step = [&](int k, float* __restrict__ cur,
                    const float* __restrict__ p1,
                    const float* __restrict__ p2) {
#pragma unroll
        for (int q = 0; q < DTW_Q; ++q) {
            const int j = t + q * DTW_T;
            const int i = k - j;
            int ic = i < 0 ? 0 : i;                    // v_med3-style clamp
            ic = ic > (DTW_N - 1) ? (DTW_N - 1) : ic;
            const float dc = fabsf(sx[ic] - syr[q]);
            const float a  = p2[j] + 2.0f * dc;
            const float m  = fminf(p1[j + 1], p1[j]);
            cur[j + 1] = fminf(a, m + dc);
        }
        __syncthreads();
    };

    // 2046 diagonals (k = 1..2N-2), trip count divisible by 3: static buffer
    // names -> DS immediate offsets, no pointer rotation.
    for (int kk = 1; kk < 2 * DTW_N - 1; kk += 3) {
        step(kk,     buf1, buf0, buf2);
        step(kk + 1, buf2, buf1, buf0);
        step(kk + 2, buf0, buf2, buf1);
    }

    // Last diagonal (k = 2N-2 = 2046) was written into buf0; corner at pos N.
    if (t == 0) {
        per_batch[b] = buf0[DTW_N] * (1.0f / (2.0f * (float)DTW_N));
    }
}

// Deterministic fixed-order reduction (no float atomics -> bitwise-stable).
__global__ void dtw_reduce_kernel(const float* __restrict__ per_batch,
                                  float* __restrict__ out, int B)
{
    if (blockIdx.x == 0 && threadIdx.x == 0) {
        float s = 0.0f;
        for (int b = 0; b < B; ++b) s += per_batch[b];
        out[0] = s / (float)B;
    }
}

extern "C" void kernel_launch(void* const* d_in, const int* in_sizes, int n_in,
                              void* d_out, int out_size, void* d_ws, size_t ws_size,
                              hipStream_t stream) {
    const float* X = (const float*)d_in[0];
    const float* Y = (const float*)d_in[1];
    float* out = (float*)d_out;
    float* ws  = (float*)d_ws;   // B floats of scratch

    const int B = in_sizes[0] / DTW_N;   // 64 for the reference shapes

    dtw_wavefront_kernel<<<B, DTW_T, 0, stream>>>(X, Y, ws);
    dtw_reduce_kernel<<<1, 32, 0, stream>>>(ws, out, B);
}